// NegLogProbLoss_90117003804744
// MI455X (gfx1250) — compile-verified
//
#include <hip/hip_runtime.h>

// ---------------------------------------------------------------------------
// Gaussian NLL mean-reduction for MI455X (gfx1250, wave32).
//   out = (1/N) * sum_i [ (t-m)^2 / v + ln v ] + ln(2*pi)
// Roofline: 3 x 65.5 MB read @ 23.3 TB/s => ~8.4 us floor; ~130 MFLOP + 33M
// trans ops are negligible -> pure streaming reduction.
// Strategy: NT b128 loads (x2 unrolled, 6 in flight), v_rcp_f32 + v_log_f32
// hardware trans, wave reduction on V_WMMA_F32_16X16X4_F32 (B = ones -> row
// sums) finished with one ds_swizzle SWAPX16.
// ---------------------------------------------------------------------------

typedef __attribute__((ext_vector_type(2))) float v2f;
typedef __attribute__((ext_vector_type(4))) float v4f;
typedef __attribute__((ext_vector_type(8))) float v8f;

#define LN2_F     0.69314718055994530942f
#define LOG_2PI_F 1.83787706640934548356f

// ds_swizzle group-of-32 encoding: and=0x1f, or=0, xor=0x10  -> lane ^= 16
#define SWIZZLE_SWAPX16 0x401F

// Reduce 32 per-lane f32 partials to the wave total using the f32 WMMA.
// A layout (16x4, 32-bit): lanes 0-15 VGPR0=K0, VGPR1=K1; lanes 16-31
// VGPR0=K2, VGPR1=K3.  With a = {x, 0} per lane and B = all-ones:
//   D[m][n] = x[m] + x[m+16]   (replicated across n)
// D layout: VGPR v holds row M=v (lanes 0-15) / M=v+8 (lanes 16-31), so
// sum(d[0..7]) gives rows 0-7 on lanes<16 and rows 8-15 on lanes>=16;
// one SWAPX16 swizzle + add completes the 32-lane sum (EXEC all ones).
__device__ __forceinline__ float wave_reduce_f32_wmma(float x) {
  v2f a; a[0] = x;    a[1] = 0.0f;
  v2f b; b[0] = 1.0f; b[1] = 1.0f;
  v8f c = {};
  c = __builtin_amdgcn_wmma_f32_16x16x4_f32(
      /*neg_a=*/false, a, /*neg_b=*/false, b,
      /*c_mod=*/(short)0, c, /*reuse_a=*/false, /*reuse_b=*/false);
  float s = ((c[0] + c[1]) + (c[2] + c[3])) + ((c[4] + c[5]) + (c[6] + c[7]));
  s += __int_as_float(
      __builtin_amdgcn_ds_swizzle(__float_as_int(s), SWIZZLE_SWAPX16));
  return s;  // every lane holds the wave total
}

__device__ __forceinline__ float nll4(v4f m, v4f v, v4f t, float acc) {
#pragma unroll
  for (int k = 0; k < 4; ++k) {
    float d  = t[k] - m[k];
    float r  = __builtin_amdgcn_rcpf(v[k]);   // v_rcp_f32
    float l2 = __builtin_amdgcn_logf(v[k]);   // v_log_f32 (log2)
    acc = __builtin_fmaf(l2, LN2_F, __builtin_fmaf(d * d, r, acc));
  }
  return acc;
}

__global__ __launch_bounds__(256) void nll_partial_kernel(
    const float* __restrict__ mean_, const float* __restrict__ var_,
    const float* __restrict__ tgt_, float* __restrict__ ws, int n4) {
  const v4f* m4 = (const v4f*)mean_;
  const v4f* v4 = (const v4f*)var_;
  const v4f* t4 = (const v4f*)tgt_;

  const int stride = gridDim.x * blockDim.x;
  int i = blockIdx.x * blockDim.x + threadIdx.x;

  float acc0 = 0.0f, acc1 = 0.0f;
  // x2 unroll: 6 independent NT b128 loads in flight, 2 FMA chains.
  for (; i + stride < n4; i += 2 * stride) {
    v4f m0 = __builtin_nontemporal_load(m4 + i);
    v4f v0 = __builtin_nontemporal_load(v4 + i);
    v4f t0 = __builtin_nontemporal_load(t4 + i);
    v4f m1 = __builtin_nontemporal_load(m4 + i + stride);
    v4f v1 = __builtin_nontemporal_load(v4 + i + stride);
    v4f t1 = __builtin_nontemporal_load(t4 + i + stride);
    acc0 = nll4(m0, v0, t0, acc0);
    acc1 = nll4(m1, v1, t1, acc1);
  }
  for (; i < n4; i += stride) {  // tail
    v4f m = __builtin_nontemporal_load(m4 + i);
    v4f v = __builtin_nontemporal_load(v4 + i);
    v4f t = __builtin_nontemporal_load(t4 + i);
    acc0 = nll4(m, v, t, acc0);
  }
  float acc = acc0 + acc1;

  // Wave reduce via f32 WMMA (all 256 threads reach here, EXEC all ones).
  float wsum = wave_reduce_f32_wmma(acc);

  __shared__ float lds[8];
  const int lane = threadIdx.x & 31;
  const int wave = threadIdx.x >> 5;
  if (lane == 0) lds[wave] = wsum;
  __syncthreads();
  if (threadIdx.x == 0) {
    float b = 0.0f;
#pragma unroll
    for (int w = 0; w < 8; ++w) b += lds[w];
    ws[blockIdx.x] = b;  // one deterministic partial per block
  }
}

__global__ __launch_bounds__(1024) void nll_final_kernel(
    const float* __restrict__ ws, float* __restrict__ out, int nparts,
    float inv_n) {
  float x = 0.0f;
  for (int i = threadIdx.x; i < nparts; i += 1024) x += ws[i];

  float wsum = wave_reduce_f32_wmma(x);

  __shared__ float lds[32];
  const int lane = threadIdx.x & 31;
  const int wave = threadIdx.x >> 5;
  if (lane == 0) lds[wave] = wsum;
  __syncthreads();
  if (wave == 0) {  // whole wave 0 active -> EXEC all ones for the WMMA
    float y   = lds[lane];  // 1024 threads / 32 = 32 wave partials
    float tot = wave_reduce_f32_wmma(y);
    if (lane == 0) out[0] = tot * inv_n + LOG_2PI_F;
  }
}

extern "C" void kernel_launch(void* const* d_in, const int* in_sizes, int n_in,
                              void* d_out, int out_size, void* d_ws,
                              size_t ws_size, hipStream_t stream) {
  const float* mean_ = (const float*)d_in[0];
  const float* var_  = (const float*)d_in[1];
  const float* tgt_  = (const float*)d_in[2];
  float* out = (float*)d_out;
  float* ws  = (float*)d_ws;

  const int n  = in_sizes[0];  // 16,384,000 (divisible by 4)
  const int n4 = n / 4;

  // 2048 blocks x 256 threads = 16384 wave32s; 3-4 unrolled (x2) iterations
  // per lane. Partials: 8 KB of workspace.
  const int BLOCKS = 2048, THREADS = 256;
  nll_partial_kernel<<<BLOCKS, THREADS, 0, stream>>>(mean_, var_, tgt_, ws, n4);

  // out = total / n + ln(2*pi)   (0.5 and the size-2 axis fold into 1/n)
  const float inv_n = 1.0f / (float)n;
  nll_final_kernel<<<1, 1024, 0, stream>>>(ws, out, BLOCKS, inv_n);
}